// GCN_88665304858771
// MI455X (gfx1250) — compile-verified
//
#include <hip/hip_runtime.h>
#include <math.h>

typedef __attribute__((ext_vector_type(2))) float v2f;
typedef __attribute__((ext_vector_type(8))) float v8f;

#define N_SRC0 100000
#define N_DST0 20000
#define N_DST1 4096
#define C_DIM  256

// ---------------------------------------------------------------- zero init
__global__ void gcn_zero_ws(float* __restrict__ ws, long n) {
  long i = (long)blockIdx.x * blockDim.x + threadIdx.x;
  long stride = (long)gridDim.x * blockDim.x;
  for (; i < n; i += stride) ws[i] = 0.0f;
}

// ---------------------------------------------------------------- degrees
__global__ void gcn_degrees(const int* __restrict__ src0, const int* __restrict__ dst0, int E0,
                            const int* __restrict__ src1, const int* __restrict__ dst1, int E1,
                            float* __restrict__ dout0, float* __restrict__ din0,
                            float* __restrict__ dout1, float* __restrict__ din1) {
  int i = blockIdx.x * blockDim.x + threadIdx.x;
  if (i < E0) {
    atomicAdd(&dout0[src0[i]], 1.0f);
    atomicAdd(&din0[dst0[i]], 1.0f);
  }
  if (i < E1) {
    atomicAdd(&dout1[src1[i]], 1.0f);
    atomicAdd(&din1[dst1[i]], 1.0f);
  }
}

// ---------------------------------------------------------------- edge SpMM
// One wave32 per edge, 8 channels per lane (C_DIM = 256).
// out[dst] += feat[src] * rsqrt(max(deg_src[src],1))
__global__ void gcn_spmm_edge(const int* __restrict__ src, const int* __restrict__ dst,
                              const float* __restrict__ deg_src,
                              const float* __restrict__ feat,
                              float* __restrict__ out, int nE) {
  int tid  = blockIdx.x * blockDim.x + threadIdx.x;
  int edge = tid >> 5;
  int lane = tid & 31;
  if (edge >= nE) return;
  int s = src[edge];
  int d = dst[edge];
  float norm = 1.0f / sqrtf(fmaxf(deg_src[s], 1.0f));
  const float4* f = (const float4*)(feat + (size_t)s * C_DIM);
  float* o = out + (size_t)d * C_DIM + lane * 8;
  float4 v0 = f[lane * 2 + 0];
  float4 v1 = f[lane * 2 + 1];
  atomicAdd(o + 0, v0.x * norm);
  atomicAdd(o + 1, v0.y * norm);
  atomicAdd(o + 2, v0.z * norm);
  atomicAdd(o + 3, v0.w * norm);
  atomicAdd(o + 4, v1.x * norm);
  atomicAdd(o + 5, v1.y * norm);
  atomicAdd(o + 6, v1.z * norm);
  atomicAdd(o + 7, v1.w * norm);
}

// ---------------------------------------------------------------- WMMA GEMM
// C[m,n] = relu?( (rowscale[m] * A[m,:]) dot B[:,n] + bias[n] )
// One wave per 16x16 output tile, v_wmma_f32_16x16x4_f32 over K in steps of 4.
// fp32 A-layout: lane row = lane&15, K-pair = 2*(lane>=16); B mirrors; C/D: VGPR r -> rows {r, r+8}.
template <int K, int N, bool RELU>
__global__ void gcn_gemm_wmma(const float* __restrict__ A,     // [M x K]
                              const float* __restrict__ degA,  // per-row degree -> scale
                              const float* __restrict__ B,     // [K x N]
                              const float* __restrict__ bias,  // [N]
                              float* __restrict__ C) {         // [M x N]
  int lane = threadIdx.x & 31;
  int wave = threadIdx.x >> 5;
  int tile = blockIdx.x * (blockDim.x >> 5) + wave;
  const int ntn = N / 16;
  int m0 = (tile / ntn) * 16;
  int n0 = (tile % ntn) * 16;
  int half = lane >> 4;   // 0: K,K+1   1: K+2,K+3
  int l    = lane & 15;

  int rowA = m0 + l;
  float scale = 1.0f / sqrtf(fmaxf(degA[rowA], 1.0f));
  const float* Ap = A + (size_t)rowA * K + 2 * half;
  const float* Bp = B + (size_t)(2 * half) * N + n0 + l;

  v8f acc = {0.0f, 0.0f, 0.0f, 0.0f, 0.0f, 0.0f, 0.0f, 0.0f};
#pragma unroll 4
  for (int k = 0; k < K; k += 4) {
    v2f a, b;
    a[0] = Ap[k]     * scale;
    a[1] = Ap[k + 1] * scale;
    b[0] = Bp[(size_t)k * N];
    b[1] = Bp[(size_t)(k + 1) * N];
    acc = __builtin_amdgcn_wmma_f32_16x16x4_f32(
        /*neg_a=*/false, a, /*neg_b=*/false, b,
        /*c_mod=*/(short)0, acc, /*reuse_a=*/false, /*reuse_b=*/false);
  }

  int col = n0 + l;
  float bv = bias[col];
#pragma unroll
  for (int r = 0; r < 8; ++r) {
    int row = m0 + r + 8 * half;
    float v = acc[r] + bv;
    if (RELU) v = fmaxf(v, 0.0f);
    C[(size_t)row * N + col] = v;
  }
}

// ---------------------------------------------------------------- launcher
extern "C" void kernel_launch(void* const* d_in, const int* in_sizes, int n_in,
                              void* d_out, int out_size, void* d_ws, size_t ws_size,
                              hipStream_t stream) {
  const float* features = (const float*)d_in[0];
  const float* W1 = (const float*)d_in[1];
  const float* b1 = (const float*)d_in[2];
  const float* W2 = (const float*)d_in[3];
  const float* b2 = (const float*)d_in[4];
  const int* src0 = (const int*)d_in[5];
  const int* dst0 = (const int*)d_in[6];
  const int* src1 = (const int*)d_in[7];
  const int* dst1 = (const int*)d_in[8];
  int E0 = in_sizes[5];
  int E1 = in_sizes[7];

  float* ws = (float*)d_ws;
  float* deg_out0 = ws;                              // 100000
  float* deg_in0  = deg_out0 + N_SRC0;               // 20000
  float* deg_out1 = deg_in0  + N_DST0;               // 20000
  float* deg_in1  = deg_out1 + N_DST0;               // 4096
  float* agg0     = deg_in1  + N_DST1;               // 20000*256
  float* agg1     = agg0 + (size_t)N_DST0 * C_DIM;   // 4096*256
  float* h1       = agg1 + (size_t)N_DST1 * C_DIM;   // 20000*256 (fully overwritten)

  long nzero = (long)(N_SRC0 + 2 * N_DST0 + N_DST1) +
               (long)N_DST0 * C_DIM + (long)N_DST1 * C_DIM;
  gcn_zero_ws<<<2048, 256, 0, stream>>>(ws, nzero);

  int Emax = (E0 > E1) ? E0 : E1;
  gcn_degrees<<<(Emax + 255) / 256, 256, 0, stream>>>(
      src0, dst0, E0, src1, dst1, E1, deg_out0, deg_in0, deg_out1, deg_in1);

  // Layer 1 aggregate: features (norm_src folded in) -> agg0 [20000 x 256]
  gcn_spmm_edge<<<(E0 * 32 + 255) / 256, 256, 0, stream>>>(
      src0, dst0, deg_out0, features, agg0, E0);

  // Layer 1 GEMM: relu((norm_dst0 * agg0) @ W1 + b1) -> h1 [20000 x 256]
  // tiles = (20000/16)*(256/16) = 20000 waves = 2500 blocks of 8 waves
  gcn_gemm_wmma<256, 256, true><<<(N_DST0 / 16) * (256 / 16) / 8, 256, 0, stream>>>(
      agg0, deg_in0, W1, b1, h1);

  // Layer 2 aggregate: h1 (norm_src1 folded in) -> agg1 [4096 x 256]
  gcn_spmm_edge<<<(E1 * 32 + 255) / 256, 256, 0, stream>>>(
      src1, dst1, deg_out1, h1, agg1, E1);

  // Layer 2 GEMM: (norm_dst1 * agg1) @ W2 + b2 -> d_out [4096 x 128]
  // tiles = (4096/16)*(128/16) = 2048 waves = 256 blocks of 8 waves
  gcn_gemm_wmma<256, 128, false><<<(N_DST1 / 16) * (128 / 16) / 8, 256, 0, stream>>>(
      agg1, deg_in1, W2, b2, (float*)d_out);
}